// RegularGraphConv_55894704390557
// MI455X (gfx1250) — compile-verified
//
#include <hip/hip_runtime.h>

// ---------------------------------------------------------------------------
// RegularGraphConv for MI455X (gfx1250):
//   out[b, :, n] = W @ gather(xT, neighbours[b,n,:]) + bias
//
// Roofline: 10.65 GFLOP (negligible vs matrix peak) vs ~72 MB compulsory HBM
// traffic; the 666 MB gather stream hits L2 (25.6 MB transposed feature table
// resident in 192 MB L2). So: FP32 WMMA (exact), transpose pre-pass for 128B
// contiguous gather rows, async global->LDS gather double-buffered against
// the WMMA reduction, non-temporal for touch-once streams.
// ---------------------------------------------------------------------------

typedef float v2f  __attribute__((ext_vector_type(2)));
typedef float v8f  __attribute__((ext_vector_type(8)));
typedef float vf4  __attribute__((ext_vector_type(4)));

constexpr int B    = 2;
constexpr int N    = 100000;
constexpr int K    = 26;
constexpr int CIN  = 32;
constexpr int COUT = 32;
constexpr int NP1  = N + 1;
constexpr int FAN  = CIN * K;            // 832
constexpr int WROW = FAN + 4;            // padded LDS row stride for W (832%64==0 would be 16-way conflict)
constexpr int AROW = 36;                 // padded LDS row stride for gathered tile
constexpr int TILES_PER_BATCH = N / 16;  // 6250
constexpr int TOTAL_TILES = B * TILES_PER_BATCH;   // 12500
constexpr int WAVES_PER_BLOCK = 4;
constexpr int SW_FLOATS = COUT * WROW;             // 26752
constexpr int SA_FLOATS = 16 * AROW;               // 576 per buffer
constexpr size_t SMEM_BYTES =
    (size_t)(SW_FLOATS + WAVES_PER_BLOCK * 2 * SA_FLOATS) * sizeof(float); // 125440 B

// ---------------------------------------------------------------------------
// Kernel 1: transpose inp [B, CIN, N] -> xT [B, N+1, CIN], append fill row.
// ---------------------------------------------------------------------------
__global__ void transpose_fill_kernel(const float* __restrict__ inp,
                                      const float* __restrict__ fill,
                                      float* __restrict__ xT) {
    __shared__ float tile[CIN][256 + 1];
    const int nblk = (N + 255) / 256;
    const int b   = blockIdx.x / nblk;
    const int t   = blockIdx.x % nblk;
    const int n0  = t * 256;
    const int tid = threadIdx.x;

    const float* src = inp + (size_t)b * CIN * N;
#pragma unroll
    for (int c = 0; c < CIN; ++c) {
        const int n = n0 + tid;
        tile[c][tid] = (n < N) ? __builtin_nontemporal_load(&src[(size_t)c * N + n]) : 0.f;
    }
    __syncthreads();

    float* dst = xT + (size_t)b * NP1 * CIN;
#pragma unroll
    for (int r = 0; r < 32; ++r) {
        const int e = r * 256 + tid;
        const int n = e >> 5;   // 0..255
        const int c = e & 31;
        if (n0 + n < N) dst[(size_t)(n0 + n) * CIN + c] = tile[c][n];
    }
    if (t == 0 && tid < CIN) dst[(size_t)N * CIN + tid] = fill[tid];
}

// ---------------------------------------------------------------------------
// Async gather of 16 feature rows (128B each) into an LDS tile buffer.
// 128 chunks of 16B; 4 GLOBAL_LOAD_ASYNC_TO_LDS_B128 per lane (ASYNCcnt).
// ---------------------------------------------------------------------------
__device__ __forceinline__ void issue_gather(const float* __restrict__ xb, int j,
                                             int lane, uint32_t buf_byte_off) {
#pragma unroll
    for (int i = 0; i < 4; ++i) {
        const int cid   = i * 32 + lane;
        const int nodeL = cid >> 3;      // 0..15
        const int part  = cid & 7;       // 16B chunk within row
        const int sj    = __shfl(j, nodeL, 32);
        const float* g  = xb + (size_t)sj * CIN + part * 4;
        const uint32_t d = buf_byte_off + (uint32_t)(nodeL * AROW + part * 4) * 4u;
        asm volatile("global_load_async_to_lds_b128 %0, %1, off"
                     :: "v"(d), "v"(g) : "memory");
    }
}

// ---------------------------------------------------------------------------
// Kernel 2: gather + FP32 WMMA GEMM. One wave per 16-node tile.
// ---------------------------------------------------------------------------
__global__ void __launch_bounds__(128, 1)
graphconv_wmma_kernel(const float* __restrict__ xT,
                      const int*   __restrict__ neigh,
                      const float* __restrict__ W,
                      const float* __restrict__ bias,
                      float* __restrict__ out) {
    extern __shared__ float smem[];
    float* sW = smem;

    const int tid  = threadIdx.x;
    const int lane = tid & 31;
    const int wave = tid >> 5;
    float* sA0 = smem + SW_FLOATS + wave * 2 * SA_FLOATS;
    float* sA1 = sA0 + SA_FLOATS;

    // ---- Stage W [32 x 832] into LDS, padded rows, no div/mod ----
    {
        const vf4* W4 = (const vf4*)W;           // 208 float4 per row
        const int o  = tid >> 2;                 // 0..31 (row)
        const int th = tid & 3;                  // 4 threads per row
#pragma unroll
        for (int i = 0; i < 52; ++i) {
            const int f4 = i * 4 + th;           // 0..207
            *(vf4*)&sW[o * WROW + f4 * 4] = W4[o * (FAN / 4) + f4];
        }
    }
    __syncthreads();

    const int tileId    = blockIdx.x * WAVES_PER_BLOCK + wave;
    const int b         = tileId / TILES_PER_BATCH;
    const int node_base = (tileId % TILES_PER_BATCH) * 16;

    const int m    = lane & 15;              // node row within tile / cout col
    const int koff = (lane < 16) ? 0 : 2;    // f32 A/B fragment K split

    const float* xb = xT + (size_t)b * NP1 * CIN;
    const int*   nb = neigh + ((size_t)b * N + node_base + m) * K;

    const float* bufs[2]   = { sA0, sA1 };
    const uint32_t boff[2] = { (uint32_t)(uintptr_t)sA0,
                               (uint32_t)(uintptr_t)sA1 };

    v8f acc0 = {};
    v8f acc1 = {};
    const float bv0 = bias[m];
    const float bv1 = bias[16 + m];

    // Prologue: start gather for neighbour 0.
    issue_gather(xb, __builtin_nontemporal_load(&nb[0]), lane, boff[0]);

    for (int k = 0; k < K; ++k) {
        const int cur = k & 1;
        if (k + 1 < K) {
            const int jn = __builtin_nontemporal_load(&nb[k + 1]);
            // WAR guard: DS reads of the next buffer (iter k-1) must finish
            // before async writes land (DScnt and ASYNCcnt are unordered).
            asm volatile("s_wait_dscnt 0x0" ::: "memory");
            issue_gather(xb, jn, lane, boff[cur ^ 1]);
            // Async loads complete in order: <=4 outstanding means iter-k's
            // 4 transfers have landed in LDS.
            asm volatile("s_wait_asynccnt 0x4" ::: "memory");
        } else {
            asm volatile("s_wait_asynccnt 0x0" ::: "memory");
        }

        const float* a_buf = bufs[cur];
        const int kf = k * CIN;
#pragma unroll
        for (int q = 0; q < 8; ++q) {
            const v2f a  = *(const v2f*)&a_buf[m * AROW + q * 4 + koff];
            const v2f b0 = *(const v2f*)&sW[m * WROW + kf + q * 4 + koff];
            const v2f b1 = *(const v2f*)&sW[(m + 16) * WROW + kf + q * 4 + koff];
            acc0 = __builtin_amdgcn_wmma_f32_16x16x4_f32(
                false, a, false, b0, (short)0, acc0, false, false);
            acc1 = __builtin_amdgcn_wmma_f32_16x16x4_f32(
                false, a, false, b1, (short)0, acc1, false, false);
        }
    }

    // ---- Epilogue: bias add, transpose C tile via LDS, NT coalesced store ----
    const int rbase = (lane < 16) ? 0 : 8;   // C layout: lanes 16-31 hold M=8..15
#pragma unroll
    for (int r = 0; r < 8; ++r) {
        sA0[(rbase + r) * 33 + m]      = acc0[r] + bv0;
        sA0[(rbase + r) * 33 + 16 + m] = acc1[r] + bv1;
    }
    float* ob = out + (size_t)b * COUT * N + node_base;
#pragma unroll
    for (int rep = 0; rep < 16; ++rep) {
        const int e  = rep * 32 + lane;
        const int o  = e >> 4;   // cout
        const int nd = e & 15;   // node
        __builtin_nontemporal_store(sA0[nd * 33 + o], &ob[(size_t)o * N + nd]);
    }
}

// ---------------------------------------------------------------------------
extern "C" void kernel_launch(void* const* d_in, const int* in_sizes, int n_in,
                              void* d_out, int out_size, void* d_ws, size_t ws_size,
                              hipStream_t stream) {
    const float* inp   = (const float*)d_in[0];
    const int*   neigh = (const int*)d_in[1];
    const float* W     = (const float*)d_in[2];
    const float* bias  = (const float*)d_in[3];
    const float* fill  = (const float*)d_in[4];
    float* outp = (float*)d_out;
    float* xT   = (float*)d_ws;   // B*(N+1)*CIN floats = 25.6 MB

    const int nblk = (N + 255) / 256;
    transpose_fill_kernel<<<B * nblk, 256, 0, stream>>>(inp, fill, xT);
    graphconv_wmma_kernel<<<TOTAL_TILES / WAVES_PER_BLOCK, 128, SMEM_BYTES, stream>>>(
        xT, neigh, W, bias, outp);
}